// GaussianConditionalStanh_45157286150660
// MI455X (gfx1250) — compile-verified
//
#include <hip/hip_runtime.h>

// ---------------------------------------------------------------------------
// GaussianConditionalStanh quantize+dequantize for MI455X (gfx1250).
// Memory-bound: 16 B/element -> ~19.4 us floor at 23.3 TB/s.
// Codebook (60 floats) staged into LDS via async global->LDS load; nearest-
// neighbor via 6-step binary search over LDS midpoints (ds_load_b32 gathers).
// Streaming traffic uses non-temporal 128-bit vmem ops.
// ---------------------------------------------------------------------------

typedef float f4 __attribute__((ext_vector_type(4)));
typedef int   i4 __attribute__((ext_vector_type(4)));

typedef __attribute__((address_space(1))) int g_i32;  // global AS int
typedef __attribute__((address_space(3))) int l_i32;  // LDS AS int

#define MAX_L 64  // codebook capacity in LDS (reference uses L=60)

#if defined(__has_builtin)
#  if __has_builtin(__builtin_amdgcn_global_load_async_to_lds_b32)
#    define HAVE_ASYNC_LDS 1
#  endif
#  if __has_builtin(__builtin_amdgcn_s_wait_asynccnt)
#    define HAVE_WAIT_ASYNCCNT 1
#  endif
#endif

// searchsorted(mid, y), side='left': first index where y <= mid[i].
__device__ __forceinline__ int search_mid(const float* s_mid, int nmid, float y) {
    int lo = 0, hi = nmid;           // answer in [0, nmid] == [0, L-1]
    while (lo < hi) {                // <= 6 iterations for L = 60
        int m = (lo + hi) >> 1;
        float v = s_mid[m];          // ds_load_b32 (per-lane divergent gather)
        if (v < y) lo = m + 1; else hi = m;
    }
    return lo;
}

__global__ __launch_bounds__(256) void stanh_quant_vec4(
    const float* __restrict__ x,
    const float* __restrict__ mu,
    const float* __restrict__ cb,
    int*   __restrict__ sym,
    float* __restrict__ deq,
    long long n4, int L)
{
    __shared__ float s_cb[MAX_L];
    __shared__ float s_mid[MAX_L];

    const int tid = threadIdx.x;

    // Stage codebook into LDS (async global->LDS copy path on CDNA5).
    if (tid < L) {
#if HAVE_ASYNC_LDS
        __builtin_amdgcn_global_load_async_to_lds_b32(
            (g_i32*)(cb + tid),
            (l_i32*)(s_cb + tid),
            /*offset=*/0, /*cpol=*/0);
#else
        s_cb[tid] = cb[tid];
#endif
    }
#if HAVE_ASYNC_LDS
#  if HAVE_WAIT_ASYNCCNT
    __builtin_amdgcn_s_wait_asynccnt(0);
#  else
    asm volatile("s_wait_asynccnt 0" ::: "memory");
#  endif
#endif
    __syncthreads();

    // Decision boundaries: mid[l] = 0.5*(cb[l] + cb[l+1]).
    if (tid < L - 1) s_mid[tid] = 0.5f * (s_cb[tid] + s_cb[tid + 1]);
    __syncthreads();

    long long i = (long long)blockIdx.x * blockDim.x + tid;
    if (i >= n4) return;

    // Streaming 128-bit non-temporal loads (global_load_b128, TH=NT).
    f4 xv = __builtin_nontemporal_load((const f4*)x + i);
    f4 mv = __builtin_nontemporal_load((const f4*)mu + i);
    f4 y  = xv - mv;

    const int nmid = L - 1;
    i4 s;
    s.x = search_mid(s_mid, nmid, y.x);
    s.y = search_mid(s_mid, nmid, y.y);
    s.z = search_mid(s_mid, nmid, y.z);
    s.w = search_mid(s_mid, nmid, y.w);

    f4 d;
    d.x = s_cb[s.x] + mv.x;   // ds_load_b32 gather + add
    d.y = s_cb[s.y] + mv.y;
    d.z = s_cb[s.z] + mv.z;
    d.w = s_cb[s.w] + mv.w;

    // Streaming 128-bit non-temporal stores.
    __builtin_nontemporal_store(s, (i4*)sym + i);
    __builtin_nontemporal_store(d, (f4*)deq + i);
}

// Scalar tail for n % 4 (not hit for the reference shape, kept for generality).
__global__ void stanh_quant_tail(
    const float* __restrict__ x, const float* __restrict__ mu,
    const float* __restrict__ cb, int* __restrict__ sym,
    float* __restrict__ deq, long long start, long long n, int L)
{
    long long i = start + threadIdx.x;
    if (i >= n) return;
    float m_ = mu[i];
    float y  = x[i] - m_;
    int lo = 0, hi = L - 1;
    while (lo < hi) {
        int m = (lo + hi) >> 1;
        float mid = 0.5f * (cb[m] + cb[m + 1]);
        if (mid < y) lo = m + 1; else hi = m;
    }
    sym[i] = lo;
    deq[i] = cb[lo] + m_;
}

extern "C" void kernel_launch(void* const* d_in, const int* in_sizes, int n_in,
                              void* d_out, int out_size, void* d_ws, size_t ws_size,
                              hipStream_t stream) {
    const float* x  = (const float*)d_in[0];   // inputs [B,C,H,W] f32
    const float* mu = (const float*)d_in[1];   // means  [B,C,H,W] f32
    const float* cb = (const float*)d_in[2];   // codebook [L] f32, sorted

    const long long n = (long long)in_sizes[0];
    const int       L = in_sizes[2];

    int*   sym = (int*)d_out;                  // symbols: first n elements (int32)
    float* deq = (float*)d_out + n;            // dequant: next n elements (f32)

    const long long n4  = n >> 2;
    const long long rem = n - (n4 << 2);

    if (n4 > 0) {
        dim3 block(256);
        dim3 grid((unsigned)((n4 + 255) / 256));
        hipLaunchKernelGGL(stanh_quant_vec4, grid, block, 0, stream,
                           x, mu, cb, sym, deq, n4, L);
    }
    if (rem > 0) {
        hipLaunchKernelGGL(stanh_quant_tail, dim3(1), dim3(32), 0, stream,
                           x, mu, cb, sym, deq, n4 << 2, n, L);
    }
}